// WindowAlignmentLayer_65876208386448
// MI455X (gfx1250) — compile-verified
//
#include <hip/hip_runtime.h>

// Sliding-window unfold: out[b, i, j] = x[b, i + j]
// B=128, L=8192, W=31, r=15, n_win=8162.
// Pure data movement; store-bandwidth bound (~129.5 MB out @ 23.3 TB/s ~ 5.7us).
// Strategy: async-stage the small reused input segment into LDS (CDNA5
// GLOBAL_LOAD_ASYNC_TO_LDS_B128 + s_wait_asynccnt), then emit guaranteed
// 16B-aligned coalesced global_store_b128 of the output fed from LDS.

#define B_      128
#define L_      8192
#define W_      31
#define R_      15
#define NWIN_   (L_ - 2 * R_)                 // 8162
#define NW_     256                           // window positions per block
#define CHUNKS_ ((NWIN_ + NW_ - 1) / NW_)     // 32
#define SEGMAX_ 288                           // 286 rounded up to x4 floats

#ifndef __has_builtin
#define __has_builtin(x) 0
#endif

#if __has_builtin(__builtin_amdgcn_global_load_async_to_lds_b128)
#define HAVE_ASYNC_B128 1
#else
#define HAVE_ASYNC_B128 0
#endif

#if __has_builtin(__builtin_amdgcn_s_wait_asynccnt)
#define HAVE_WAIT_ASYNC 1
#else
#define HAVE_WAIT_ASYNC 0
#endif

// 128-bit payload type in the address spaces the async builtin expects:
// param 0: int4 in AS(1) (global), param 1: AS(3) (LDS).
typedef int b128_t __attribute__((ext_vector_type(4)));
typedef __attribute__((address_space(1))) b128_t GB128;
typedef __attribute__((address_space(3))) b128_t LB128;

__global__ __launch_bounds__(256) void unfold_w31_kernel(
    const float* __restrict__ x, float* __restrict__ out) {
  __shared__ float s_seg[SEGMAX_];

  const int blk   = blockIdx.x;
  const int b     = blk >> 5;          // / CHUNKS_ (=32)
  const int chunk = blk & (CHUNKS_ - 1);
  const int i0    = chunk * NW_;
  const int cnt   = (NWIN_ - i0) < NW_ ? (NWIN_ - i0) : NW_;   // 256 or 226
  const int seg_len = cnt + 2 * R_;                            // 286 or 256
  const int tid   = threadIdx.x;

  const float* src = x + (size_t)b * L_ + i0;   // 1024-byte aligned

#if HAVE_ASYNC_B128
  {
    // lane-wise async b128 copies: 72 lanes (full chunk) or 64 (last chunk).
    // Full chunk reads 288 floats; i0 <= 7680 so i0+288 <= 7968 < L_ (in-row).
    const int nv16 = (seg_len + 3) >> 2;
    if (tid < nv16) {
      __builtin_amdgcn_global_load_async_to_lds_b128(
          (GB128*)(src + tid * 4), (LB128*)(&s_seg[tid * 4]),
          /*offset=*/0, /*cpol=*/0);
    }
#if HAVE_WAIT_ASYNC
    __builtin_amdgcn_s_wait_asynccnt(0);
#else
    asm volatile("s_wait_asynccnt 0" ::: "memory");
#endif
    __syncthreads();
  }
#else
  {
    for (int t = tid; t < seg_len; t += 256) s_seg[t] = src[t];
    __syncthreads();
  }
#endif

  float* outp = out + ((size_t)b * NWIN_ + i0) * W_;
  const int total = cnt * W_;     // 7936 or 7006

  // Alignment prolog: block base is 0 or 8 bytes off a 16B boundary
  // (base floats = b*253022 + i0*31; mod 4 == (2b) mod 4). lead in {0, 2}.
  const int mis  = (int)(((uintptr_t)outp >> 2) & 3);   // in floats
  const int lead = (4 - mis) & 3;
  if (tid < lead) {
    // p < 31  =>  iw = 0, jw = p  =>  value = s_seg[p]
    outp[tid] = s_seg[tid];
  }

  // Main body: 16B-aligned float4 stores -> global_store_b128 guaranteed.
  const int nvec = (total - lead) >> 2;
  for (int c = tid; c < nvec; c += 256) {
    const int p = lead + (c << 2);
    int iw = p / W_;              // const division -> mul-hi
    int jw = p - iw * W_;
    float4 v;
    v.x = s_seg[iw + jw];
    if (++jw == W_) { jw = 0; ++iw; }
    v.y = s_seg[iw + jw];
    if (++jw == W_) { jw = 0; ++iw; }
    v.z = s_seg[iw + jw];
    if (++jw == W_) { jw = 0; ++iw; }
    v.w = s_seg[iw + jw];
    *reinterpret_cast<float4*>(outp + p) = v;   // aligned b128 store
  }

  // Scalar tail (< 4 elements)
  for (int p = lead + (nvec << 2) + tid; p < total; p += 256) {
    const int iw = p / W_;
    const int jw = p - iw * W_;
    outp[p] = s_seg[iw + jw];
  }
}

extern "C" void kernel_launch(void* const* d_in, const int* in_sizes, int n_in,
                              void* d_out, int out_size, void* d_ws, size_t ws_size,
                              hipStream_t stream) {
  (void)in_sizes; (void)n_in; (void)d_ws; (void)ws_size; (void)out_size;
  const float* x = (const float*)d_in[0];
  // d_in[1] is win_size (==31), baked into the kernel as a compile-time const.
  float* out = (float*)d_out;

  dim3 grid(B_ * CHUNKS_);   // 4096 blocks
  dim3 block(256);           // 8 waves (wave32)
  unfold_w31_kernel<<<grid, block, 0, stream>>>(x, out);
}